// MRConv2d_8031588843841
// MI455X (gfx1250) — compile-verified
//
#include <hip/hip_runtime.h>
#include <hip/hip_bf16.h>
#include <math.h>

typedef __attribute__((ext_vector_type(2))) float v2f;
typedef __attribute__((ext_vector_type(8))) float v8f;

namespace {
constexpr int Bc = 8, Cc = 192, Nc = 3136, Kc = 9;
constexpr int OUTc = 384, GROUPSc = 4;
constexpr int CPG = Cc / GROUPSc;   // 48 original channels per group
constexpr int CG  = 2 * CPG;        // 96 interleaved in-channels per group
constexpr int OG  = OUTc / GROUPSc; // 96 out-channels per group
constexpr int NT  = 64;             // N tile (3136 = 49 * 64, all tiles full)
constexpr int NTILES = Nc / NT;     // 49
constexpr float BN_EPS = 1e-5f;
}

// One block = (batch b, group g, 64-wide n-tile). Fuses the max-relative
// gather + channel interleave into the LDS fill, then does the 96x96x64
// grouped-conv GEMM with V_WMMA_F32_16X16X4_F32.
__global__ __launch_bounds__(256) void mrconv_gemm_wmma(
    const float* __restrict__ x,     // [B, C, N]
    const int*   __restrict__ edge,  // [2, B, N, K]
    const float* __restrict__ w,     // [OUT, 96] row-major
    float*       __restrict__ y)     // [B, OUT, N]
{
  __shared__ float sW[OG * CG];   // 36 KB: W_g, row-major [96][96]
  __shared__ float sX[CG * NT];   // 24 KB: interleaved [96][64]
  __shared__ int   sJ[Kc * NT];   // gather indices (x_j source)
  __shared__ int   sI[Kc * NT];   // gather indices (x_i source)

  const int blk = blockIdx.x;
  const int nt  = blk % NTILES;
  const int g   = (blk / NTILES) % GROUPSc;
  const int b   = blk / (NTILES * GROUPSc);
  const int n0  = nt * NT;
  const int t   = threadIdx.x;

  // Stage W_g into LDS (9216 floats / 256 threads = 36 each).
  const float* wg = w + (size_t)g * OG * CG;
  for (int i = t; i < OG * CG; i += 256) sW[i] = wg[i];

  // Stage edge indices, transposed to [k][n_local] for conflict-free reads.
  const int* ej = edge + ((size_t)b * Nc + n0) * Kc;        // edge_index[0] -> x_j
  const int* ei = edge + ((size_t)(Bc + b) * Nc + n0) * Kc; // edge_index[1] -> x_i
  for (int i = t; i < Kc * NT; i += 256) {
    const int n_l = i / Kc, k = i % Kc;
    sJ[k * NT + n_l] = ej[n_l * Kc + k];
    sI[k * NT + n_l] = ei[n_l * Kc + k];
  }
  __syncthreads();

  // Fill X tile: row 2*c_l = x, row 2*c_l+1 = max_k(x_j - x_i). x is fully
  // L2-resident (19 MB << 192 MB), so the random gathers stay on-chip.
  for (int i = t; i < CPG * NT; i += 256) {   // 48*64 = 3072
    const int c_l = i / NT, n_l = i % NT;
    const float* xb = x + ((size_t)b * Cc + (g * CPG + c_l)) * Nc;
    __builtin_prefetch(xb, 0, 0);             // global_prefetch_b8
    const float val = xb[n0 + n_l];
    float rel = -3.402823466e38f;
    #pragma unroll
    for (int k = 0; k < Kc; ++k)
      rel = fmaxf(rel, xb[sJ[k * NT + n_l]] - xb[sI[k * NT + n_l]]);
    sX[(2 * c_l)     * NT + n_l] = val;
    sX[(2 * c_l + 1) * NT + n_l] = rel;
  }
  __syncthreads();

  // 6 M-tiles x 4 N-tiles = 24 WMMA tiles; 8 waves -> 3 tiles per wave.
  const int wave = t >> 5, lane = t & 31;
  const int half = lane >> 4;   // 0: lanes 0-15 (K 0/1), 1: lanes 16-31 (K 2/3)
  const int l16  = lane & 15;

  v8f acc[3];
  #pragma unroll
  for (int i = 0; i < 3; ++i) acc[i] = (v8f){0, 0, 0, 0, 0, 0, 0, 0};

  for (int ks = 0; ks < CG / 4; ++ks) {       // 24 K-steps of 4
    const int k0 = ks * 4 + 2 * half;
    #pragma unroll
    for (int i = 0; i < 3; ++i) {
      const int tid = wave + 8 * i;           // 0..23, each tile exactly once
      const int mi = tid % 6, ni = tid / 6;
      // A frag: 16x4 f32, lane holds M = mi*16+l16, K = k0, k0+1 (contiguous).
      const float* ap = &sW[(mi * 16 + l16) * CG + k0];
      v2f a = { ap[0], ap[1] };
      // B frag: 4x16 f32, lane holds N = ni*16+l16, K = k0, k0+1 (stride NT).
      const float* bp = &sX[k0 * NT + ni * 16 + l16];
      v2f bb = { bp[0], bp[NT] };
      acc[i] = __builtin_amdgcn_wmma_f32_16x16x4_f32(
          false, a, false, bb, (short)0, acc[i], false, false);
    }
  }

  // C/D layout: VGPR r -> M = mi*16 + r + half*8, N = ni*16 + l16.
  #pragma unroll
  for (int i = 0; i < 3; ++i) {
    const int tid = wave + 8 * i;
    const int mi = tid % 6, ni = tid / 6;
    const int col  = n0 + ni * 16 + l16;
    const int row0 = g * OG + mi * 16 + half * 8;
    float* yp = y + ((size_t)b * OUTc + row0) * Nc + col;
    #pragma unroll
    for (int r = 0; r < 8; ++r) yp[(size_t)r * Nc] = acc[i][r];
  }
}

// Per-channel batch statistics over (B, N): one block per channel.
__global__ __launch_bounds__(256) void bn_stats(
    const float* __restrict__ y, float* __restrict__ stats)
{
  const int ch = blockIdx.x;
  float s = 0.f, ss = 0.f;
  for (int i = threadIdx.x; i < Bc * Nc; i += 256) {
    const int b = i / Nc, n = i % Nc;
    const float v = y[((size_t)b * OUTc + ch) * Nc + n];
    s += v; ss += v * v;
  }
  __shared__ float rs[256], rss[256];
  rs[threadIdx.x] = s; rss[threadIdx.x] = ss;
  __syncthreads();
  for (int off = 128; off > 0; off >>= 1) {
    if (threadIdx.x < off) {
      rs[threadIdx.x]  += rs[threadIdx.x + off];
      rss[threadIdx.x] += rss[threadIdx.x + off];
    }
    __syncthreads();
  }
  if (threadIdx.x == 0) {
    const float inv_n = 1.0f / (float)(Bc * Nc);
    const float mean = rs[0] * inv_n;
    const float var  = rss[0] * inv_n - mean * mean;  // population var (jnp.var)
    stats[ch]        = mean;
    stats[OUTc + ch] = var;
  }
}

// In-place normalize + exact-erf GELU. conv bias cancels under BN (shifts the
// mean by the same constant), so it is intentionally omitted.
__global__ __launch_bounds__(256) void bn_gelu(
    float* __restrict__ y, const float* __restrict__ stats,
    const float* __restrict__ gamma, const float* __restrict__ beta)
{
  const size_t i = (size_t)blockIdx.x * 256 + threadIdx.x;
  const int ch = (int)((i / Nc) % OUTc);
  const float mean = stats[ch];
  const float var  = stats[OUTc + ch];
  float v = (y[i] - mean) * rsqrtf(var + BN_EPS) * gamma[ch] + beta[ch];
  y[i] = 0.5f * v * (1.0f + erff(v * 0.70710678118654752f));
}

extern "C" void kernel_launch(void* const* d_in, const int* in_sizes, int n_in,
                              void* d_out, int out_size, void* d_ws, size_t ws_size,
                              hipStream_t stream) {
  const float* x      = (const float*)d_in[0];
  const int*   edge   = (const int*)d_in[1];
  const float* conv_w = (const float*)d_in[2];
  // d_in[3] = conv_b: cancels exactly under training-mode BatchNorm.
  const float* gamma  = (const float*)d_in[4];
  const float* beta   = (const float*)d_in[5];
  float* y     = (float*)d_out;           // [B, OUT, N] fp32
  float* stats = (float*)d_ws;            // 2*OUT floats (fully overwritten)

  mrconv_gemm_wmma<<<Bc * GROUPSc * NTILES, 256, 0, stream>>>(x, edge, conv_w, y);
  bn_stats<<<OUTc, 256, 0, stream>>>(y, stats);
  const int total = Bc * OUTc * Nc;       // 9,633,792 = 37632 * 256 exactly
  bn_gelu<<<total / 256, 256, 0, stream>>>(y, stats, gamma, beta);
}